// ProxyGroupLoss_77644418777843
// MI455X (gfx1250) — compile-verified
//
#include <hip/hip_runtime.h>
#include <cstdint>

// ---------------- problem constants (match reference) ----------------
#define DEGREE     31
#define THRESHOLD  16
#define COST       0.25f
#define TILE_ROWS  256         // nodes per tile == blockDim.x
#define NBLOCKS    1024

typedef float    v2f  __attribute__((ext_vector_type(2)));
typedef float    v8f  __attribute__((ext_vector_type(8)));
typedef uint32_t u32x4 __attribute__((ext_vector_type(4)));
typedef int      i32x4 __attribute__((ext_vector_type(4)));
typedef int      i32x8 __attribute__((ext_vector_type(8)));

#define HAVE_TDM       __has_builtin(__builtin_amdgcn_tensor_load_to_lds)
#define HAVE_WMMA_F32  __has_builtin(__builtin_amdgcn_wmma_f32_16x16x4_f32)

// ---- compile-time path diagnostics (show up in hipcc stderr) ----
#if HAVE_TDM
#pragma message("CDNA5-PATH: tensor_load_to_lds builtin AVAILABLE (TDM pipeline active)")
#else
#pragma message("CDNA5-PATH: tensor_load_to_lds builtin MISSING (cooperative-load fallback)")
#endif
#if HAVE_WMMA_F32
#pragma message("CDNA5-PATH: wmma_f32_16x16x4_f32 builtin AVAILABLE")
#else
#pragma message("CDNA5-PATH: wmma_f32_16x16x4_f32 builtin MISSING (shuffle fallback)")
#endif
#if defined(__clang_major__) && __clang_major__ >= 23
#pragma message("CDNA5-PATH: clang >= 23 (6-arg TDM builtin form)")
#else
#pragma message("CDNA5-PATH: clang < 23 (5-arg TDM builtin form)")
#endif

__device__ __forceinline__ float sigmoid_fast(float x) {
    // 1 / (1 + e^-x); v_exp_f32 + v_rcp_f32 (approx, ~1 ulp — fine for tolerance)
    return __builtin_amdgcn_rcpf(1.0f + __expf(-x));
}

// ---------------- exact f32 wave(32) sum via V_WMMA_F32_16X16X4_F32 ----------------
// A(16x4) holds the 32 per-lane accumulators (lane L -> VGPR0 slot), B = ones.
// D[m][n] = acc_m + acc_{m+16} replicated across n; each lane sums its 8 row
// VGPRs, then one xor-16 shuffle combines the two half-wave row groups.
__device__ __forceinline__ float wave_sum(float x) {
#if HAVE_WMMA_F32
    v2f a; a.x = x;    a.y = 0.0f;
    v2f b; b.x = 1.0f; b.y = 1.0f;
    v8f c = {};
    v8f d = __builtin_amdgcn_wmma_f32_16x16x4_f32(
        /*neg_a=*/false, a, /*neg_b=*/false, b,
        /*c_mod=*/(short)0, c, /*reuse_a=*/false, /*reuse_b=*/false);
    float s = ((d[0] + d[1]) + (d[2] + d[3])) + ((d[4] + d[5]) + (d[6] + d[7]));
    s += __shfl_xor(s, 16, 32);
    return s;
#else
    #pragma unroll
    for (int off = 16; off > 0; off >>= 1) x += __shfl_xor(x, off, 32);
    return x;
#endif
}

// ---------------- TDM: 2D tile of the neighbors matrix -> LDS ----------------
#if HAVE_TDM
__device__ __forceinline__ void tdm_issue_tile(uint32_t lds_off,
                                               const int* gsrc,
                                               uint32_t rows_remaining) {
    uint64_t ga = (uint64_t)(uintptr_t)gsrc;
    // D# group 0: count=1 | lds_addr | global_addr[56:0] | type=2
    u32x4 g0 = { 1u,
                 lds_off,
                 (uint32_t)ga,
                 (((uint32_t)(ga >> 32)) & 0x01FFFFFFu) | (2u << 30) };
    // D# group 1: data_size=4B, tensor_dim0=31, tensor_dim1=rows_remaining,
    //             tile_dim0=31, tile_dim1=256, dim0_stride=31 (contiguous rows)
    i32x8 g1 = { (int)(2u << 16),                                    // data_size=2 (4B)
                 (int)((uint32_t)DEGREE << 16),                      // tensor_dim0 lo16
                 (int)((rows_remaining & 0xFFFFu) << 16),            // tensor_dim1 lo16
                 (int)(((rows_remaining >> 16) & 0xFFFFu)
                       | ((uint32_t)DEGREE << 16)),                  // dim1 hi16 | tile_dim0
                 (int)TILE_ROWS,                                     // tile_dim1
                 (int)DEGREE,                                        // tensor_dim0_stride lo32
                 0, 0 };
    i32x4 z4 = { 0, 0, 0, 0 };
#if defined(__clang_major__) && __clang_major__ >= 23
    i32x8 z8 = { 0, 0, 0, 0, 0, 0, 0, 0 };
    __builtin_amdgcn_tensor_load_to_lds(g0, g1, z4, z4, z8, 0);
#else
    __builtin_amdgcn_tensor_load_to_lds(g0, g1, z4, z4, 0);
#endif
}
#endif

// ---------------- per-node Poisson-binomial DP (absorbing at THRESHOLD) ----------------
__device__ __forceinline__ float compute_node(const float* __restrict__ gains,
                                              const uint32_t* __restrict__ row,
                                              uint32_t node, int n) {
    if (node >= (uint32_t)n) return 0.0f;

    // prefetch all gather indices, then all gathered gains (hide L2 latency)
    uint32_t idx[DEGREE];
    #pragma unroll
    for (int j = 0; j < DEGREE; ++j) idx[j] = row[j];
    float gg[DEGREE];
    #pragma unroll
    for (int j = 0; j < DEGREE; ++j) {
        __builtin_assume(idx[j] < (1u << 24));   // enables saddr+scaled-voffset gathers
        gg[j] = gains[idx[j]];
    }
    float p0 = sigmoid_fast(gains[node]);

    // dp[k] = P(count == k), k<16; dp[16] = P(count >= 16)  (counts are monotone)
    float dp[THRESHOLD + 1];
    dp[0] = 1.0f;
    #pragma unroll
    for (int k = 1; k <= THRESHOLD; ++k) dp[k] = 0.0f;

    #pragma unroll
    for (int s = 0; s < DEGREE + 1; ++s) {
        float pj = (s == 0) ? p0 : sigmoid_fast(gg[s - 1]);
        const int mk = (s + 1 < THRESHOLD) ? (s + 1) : THRESHOLD;   // triangular trim
        if (mk == THRESHOLD)
            dp[THRESHOLD] = fmaf(dp[THRESHOLD - 1], pj, dp[THRESHOLD]); // absorb
        #pragma unroll
        for (int k = (mk < THRESHOLD ? mk : THRESHOLD - 1); k >= 1; --k)
            dp[k] = fmaf(pj, dp[k - 1] - dp[k], dp[k]);  // dp(1-p) + dp_prev*p
        dp[0] *= (1.0f - pj);
    }
    return dp[THRESHOLD] - COST * p0;
}

// ---------------- main kernel: TDM double-buffered tile pipeline ----------------
__global__ __launch_bounds__(TILE_ROWS) void pgl_main(
        const float* __restrict__ gains, const int* __restrict__ neighbors,
        float* __restrict__ partials, int atomic_mode, float* __restrict__ out, int n) {
    __shared__ __align__(16) uint32_t buf[2][TILE_ROWS * DEGREE];
    __shared__ float wsum[TILE_ROWS / 32];

    const int tid    = threadIdx.x;
    const int nTiles = (n + TILE_ROWS - 1) / TILE_ROWS;
    const int count  = ((int)blockIdx.x < nTiles)
                         ? ((nTiles - 1 - (int)blockIdx.x) / (int)gridDim.x + 1) : 0;
    float acc = 0.0f;

#if HAVE_TDM
    if (count > 0 && tid == 0) {
        uint32_t s0 = (uint32_t)blockIdx.x * TILE_ROWS;
        tdm_issue_tile((uint32_t)(uintptr_t)&buf[0][0],
                       neighbors + (size_t)s0 * DEGREE, (uint32_t)n - s0);
    }
    for (int i = 0; i < count; ++i) {
        const int      t     = (int)blockIdx.x + i * (int)gridDim.x;
        const uint32_t start = (uint32_t)t * TILE_ROWS;
        if (tid == 0) {
            if (i + 1 < count) {   // prefetch next tile into the other buffer
                const int      tn = (int)blockIdx.x + (i + 1) * (int)gridDim.x;
                const uint32_t sn = (uint32_t)tn * TILE_ROWS;
                tdm_issue_tile((uint32_t)(uintptr_t)&buf[(i + 1) & 1][0],
                               neighbors + (size_t)sn * DEGREE, (uint32_t)n - sn);
                __builtin_amdgcn_s_wait_tensorcnt(1);   // oldest (current) tile done
            } else {
                __builtin_amdgcn_s_wait_tensorcnt(0);
            }
        }
        __syncthreads();                                  // tile visible to all waves
        acc += compute_node(gains, &buf[i & 1][tid * DEGREE], start + tid, n);
        __syncthreads();                                  // buffer free before next TDM write
    }
#else
    for (int i = 0; i < count; ++i) {
        const int      t     = (int)blockIdx.x + i * (int)gridDim.x;
        const uint32_t start = (uint32_t)t * TILE_ROWS;
        const int*     src   = neighbors + (size_t)start * DEGREE;
        const int rows_rem   = n - (int)start;
        const int valid_dw   = ((rows_rem < TILE_ROWS) ? rows_rem : TILE_ROWS) * DEGREE;
        for (int k = tid; k < TILE_ROWS * DEGREE; k += TILE_ROWS)
            buf[i & 1][k] = (k < valid_dw) ? (uint32_t)src[k] : 0u;
        __syncthreads();
        acc += compute_node(gains, &buf[i & 1][tid * DEGREE], start + tid, n);
        __syncthreads();
    }
#endif

    // exact f32 block reduction: WMMA wave sums -> LDS -> thread 0
    float s = wave_sum(acc);
    if ((tid & 31) == 0) wsum[tid >> 5] = s;
    __syncthreads();
    if (tid == 0) {
        float bs = 0.0f;
        #pragma unroll
        for (int w = 0; w < TILE_ROWS / 32; ++w) bs += wsum[w];
        if (atomic_mode) atomicAdd(out, -bs);
        else             partials[blockIdx.x] = bs;
    }
}

// ---------------- deterministic final reduce (single block) ----------------
__global__ __launch_bounds__(128) void pgl_reduce(const float* __restrict__ partials,
                                                  int nparts, float* __restrict__ out) {
    __shared__ float wsum[4];
    float acc = 0.0f;
    for (int i = threadIdx.x; i < nparts; i += 128) acc += partials[i];  // fixed order
    float s = wave_sum(acc);
    if ((threadIdx.x & 31) == 0) wsum[threadIdx.x >> 5] = s;
    __syncthreads();
    if (threadIdx.x == 0) out[0] = -((wsum[0] + wsum[1]) + (wsum[2] + wsum[3]));
}

__global__ void pgl_zero(float* out) {
    if (threadIdx.x == 0 && blockIdx.x == 0) out[0] = 0.0f;
}

// ---------------- launcher ----------------
extern "C" void kernel_launch(void* const* d_in, const int* in_sizes, int n_in,
                              void* d_out, int out_size, void* d_ws, size_t ws_size,
                              hipStream_t stream) {
    const float* gains     = (const float*)d_in[0];
    const int*   neighbors = (const int*)d_in[1];   // jnp.int64 w/o x64 => int32
    float*       out       = (float*)d_out;
    const int    n         = in_sizes[0];

    const int nTiles  = (n + TILE_ROWS - 1) / TILE_ROWS;
    int       nblocks = NBLOCKS;
    if (nblocks > nTiles) nblocks = nTiles;

    if (ws_size >= (size_t)nblocks * sizeof(float)) {
        float* partials = (float*)d_ws;
        pgl_main<<<nblocks, TILE_ROWS, 0, stream>>>(gains, neighbors, partials, 0, out, n);
        pgl_reduce<<<1, 128, 0, stream>>>(partials, nblocks, out);
    } else {
        pgl_zero<<<1, 64, 0, stream>>>(out);
        pgl_main<<<nblocks, TILE_ROWS, 0, stream>>>(gains, neighbors, nullptr, 1, out, n);
    }
}